// TF_physical_layer_13365938225469
// MI455X (gfx1250) — compile-verified
//
#include <hip/hip_runtime.h>
#include <hip/hip_bf16.h>

#define BLOCK 256
#define TILE_ROWS 1024      // rows of obs_pos per LDS tile (8 KB)
#define SENTINEL 0x7FFFFFFF

// ---- CDNA5 async global->LDS path (guarded so it compiles on both toolchains)
#if defined(__gfx1250__) && __has_builtin(__builtin_amdgcn_global_load_async_to_lds_b64) && \
    __has_builtin(__builtin_amdgcn_global_load_async_to_lds_b128)
#define USE_ASYNC_LDS 1
#else
#define USE_ASYNC_LDS 0
#endif

#if USE_ASYNC_LDS
typedef int v2i __attribute__((ext_vector_type(2)));
typedef int v4i __attribute__((ext_vector_type(4)));
typedef __attribute__((address_space(1))) v2i* gv2i_p;
typedef __attribute__((address_space(3))) v2i* lv2i_p;
typedef __attribute__((address_space(1))) v4i* gv4i_p;
typedef __attribute__((address_space(3))) v4i* lv4i_p;

__device__ __forceinline__ void async_copy_b64(const void* g, void* l) {
  __builtin_amdgcn_global_load_async_to_lds_b64((gv2i_p)g, (lv2i_p)l, 0, 0);
}
__device__ __forceinline__ void async_copy_b128(const void* g, void* l) {
  __builtin_amdgcn_global_load_async_to_lds_b128((gv4i_p)g, (lv4i_p)l, 0, 0);
}
__device__ __forceinline__ void async_wait0() {
#if __has_builtin(__builtin_amdgcn_s_wait_asynccnt)
  __builtin_amdgcn_s_wait_asynccnt(0);
#else
  asm volatile("s_wait_asynccnt 0" ::: "memory");
#endif
}
#endif

// ---------------------------------------------------------------- init
__global__ void init_idx_kernel(int* __restrict__ best, int B) {
  int i = blockIdx.x * blockDim.x + threadIdx.x;
  if (i < B) best[i] = SENTINEL;
}

// ---------------------------------------------------------------- search
// grid.x: chunks of 256 query positions (one query per lane, wave32-safe)
// grid.y: row chunks of obs_pos; each block scans [row_begin, row_end)
// Tiles are staged into LDS via the gfx1250 async copy engine (b128 DMA,
// ASYNCcnt-tracked), then each lane scans two rows per float4 LDS broadcast
// read against its private (px, py). First match in ascending row order ==
// min row index -> deterministic atomicMin combine across row chunks.
__global__ __launch_bounds__(BLOCK) void star_search_kernel(
    const float* __restrict__ positions,
    const float2* __restrict__ obs,
    int* __restrict__ best,
    int B, int N, int slice) {
  __shared__ float4 tile4[TILE_ROWS / 2];
  float2* tile2 = (float2*)tile4;

  const int tid = threadIdx.x;
  const int p = blockIdx.x * BLOCK + tid;

  float px, py;
  if (p < B) {
    px = positions[2 * p];
    py = positions[2 * p + 1];
  } else {
    px = __int_as_float(0x7fc00000);  // NaN: never compares equal
    py = px;
  }

  const int row_begin = blockIdx.y * slice;
  const int row_end = min(row_begin + slice, N);
  int found = SENTINEL;

  for (int base = row_begin; base < row_end; base += TILE_ROWS) {
    const int rows = min(TILE_ROWS, row_end - base);
    const int pairs = rows >> 1;

    // ---- stage tile into LDS (2 rows per 16B chunk)
#if USE_ASYNC_LDS
    for (int c = tid; c < pairs; c += BLOCK)
      async_copy_b128((const void*)(obs + base + 2 * c), (void*)&tile4[c]);
    if ((rows & 1) && tid == 0)  // odd tail row (only possible on last tile)
      async_copy_b64((const void*)(obs + base + rows - 1), (void*)&tile2[rows - 1]);
    async_wait0();
#else
    for (int c = tid; c < pairs; c += BLOCK)
      tile4[c] = ((const float4*)(obs + base))[c];
    if ((rows & 1) && tid == 0)
      tile2[rows - 1] = obs[base + rows - 1];
#endif
    // hint the next tile toward the caches while we scan this one
    if (base + TILE_ROWS < row_end)
      __builtin_prefetch((const void*)(obs + base + TILE_ROWS + tid), 0, 0);

    __syncthreads();

    // ---- scan: 2 rows per iteration, LDS broadcast reads
    if (found == SENTINEL) {  // once matched, the min can't improve (ascending)
      for (int k = 0; k < pairs; ++k) {
        float4 v = tile4[k];
        if (v.x == px || v.y == py) { found = base + 2 * k;     break; }
        if (v.z == px || v.w == py) { found = base + 2 * k + 1; break; }
      }
      if (found == SENTINEL && (rows & 1)) {
        float2 o = tile2[rows - 1];
        if (o.x == px || o.y == py) found = base + rows - 1;
      }
    }
    __syncthreads();  // protect tile from next iteration's overwrite
  }

  if (p < B && found != SENTINEL) atomicMin(&best[p], found);
}

// ---------------------------------------------------------------- gather
__global__ void gather_kernel(const int* __restrict__ best,
                              const float* __restrict__ zks,
                              float* __restrict__ out,
                              int B, int N, int NZ, int total) {
  int i = blockIdx.x * blockDim.x + threadIdx.x;
  if (i >= total) return;
  int b = i / NZ;
  int z = i - b * NZ;
  int r = best[b];
  if (r >= N) r = 0;  // no match anywhere -> argmax(all False) == 0
  out[i] = zks[r * NZ + z];
}

// ---------------------------------------------------------------- launch
extern "C" void kernel_launch(void* const* d_in, const int* in_sizes, int n_in,
                              void* d_out, int out_size, void* d_ws, size_t ws_size,
                              hipStream_t stream) {
  const float* positions = (const float*)d_in[0];  // (B, 2)
  const float* obs_pos   = (const float*)d_in[1];  // (N, 2)
  const float* zks_prior = (const float*)d_in[2];  // (N, NZ)

  const int B  = in_sizes[0] / 2;
  const int N  = in_sizes[1] / 2;
  const int NZ = in_sizes[2] / N;

  int* best = (int*)d_ws;  // B ints of scratch

  // 1) init first-match indices to sentinel
  init_idx_kernel<<<(B + BLOCK - 1) / BLOCK, BLOCK, 0, stream>>>(best, B);

  // 2) search: 8 position-chunks x 40 row-chunks = 320 blocks
  const int row_chunks = 40;
  const int slice = (N + row_chunks - 1) / row_chunks;
  dim3 grid((B + BLOCK - 1) / BLOCK, row_chunks);
  star_search_kernel<<<grid, BLOCK, 0, stream>>>(
      positions, (const float2*)obs_pos, best, B, N, slice);

  // 3) gather zernike rows
  const int total = B * NZ;
  gather_kernel<<<(total + BLOCK - 1) / BLOCK, BLOCK, 0, stream>>>(
      best, zks_prior, (float*)d_out, B, N, NZ, total);
}